// BC_evidence_X_Estimation_56358560858220
// MI455X (gfx1250) — compile-verified
//
#include <hip/hip_runtime.h>
#include <hip/hip_bf16.h>
#include <stdint.h>

// Problem constants (from reference)
#define MU_C    0.01f
#define RHO_C   0.05f
#define SHARP_C 32.0f

// The async-to-LDS b32 builtin is typed over int* in explicit address spaces
// (AS1 = global, AS3 = LDS) per the round-1 diagnostic.
typedef __attribute__((address_space(1))) int gint_t;   // global int
typedef __attribute__((address_space(3))) int lint_t;   // LDS int

__device__ __forceinline__ float fsig(float x) {
  // sigmoid via hardware v_exp + v_rcp (two transcendental ops, ~1 ulp);
  // avoids the 10-op IEEE divide expansion seen in round 2 asm.
  return __builtin_amdgcn_rcpf(1.0f + __expf(-x));
}

// X0 = sigmoid(logit_X0) written directly into output row 0
__global__ __launch_bounds__(256) void init_sigmoid_kernel(
    const float* __restrict__ logit, float* __restrict__ x0, int n) {
  int i = blockIdx.x * 256 + threadIdx.x;
  if (i < n) x0[i] = fsig(logit[i]);
}

// One timestep: stage edge tile (u,v,s) into LDS via CDNA5 async copy,
// gather state, emit kappa, scatter symmetric diffusion update into Xn.
__global__ __launch_bounds__(256) void edge_kernel(
    const float* __restrict__ Xt,      // state at time t (read)
    float* __restrict__ Xn,            // state at time t+1 (atomic accumulate); unused if !doUpd
    const int* __restrict__ u,
    const int* __restrict__ v,
    const int* __restrict__ s,
    const float* __restrict__ theta,   // 1 element
    float* __restrict__ kap,           // kappa row for time t
    int E, int doUpd) {
  __shared__ int su[256];
  __shared__ int sv[256];
  __shared__ int ss[256];

  const int tid = threadIdx.x;
  const int e   = blockIdx.x * 256 + tid;

  // --- CDNA5 async global -> LDS staging of the streamed edge arrays ---
  // Per-lane 4B copies, tracked by ASYNCcnt; inactive lanes (tail block) are
  // EXEC-masked out and simply don't transfer.
  if (e < E) {
    __builtin_amdgcn_global_load_async_to_lds_b32((gint_t*)(u + e), (lint_t*)(su + tid), 0, 0);
    __builtin_amdgcn_global_load_async_to_lds_b32((gint_t*)(v + e), (lint_t*)(sv + tid), 0, 0);
    __builtin_amdgcn_global_load_async_to_lds_b32((gint_t*)(s + e), (lint_t*)(ss + tid), 0, 0);
  }

  // Overlap: compute eps while the async copies are in flight (uniform scalar).
  const float eps = 0.5f * fsig(theta[0]);

  __builtin_amdgcn_s_wait_asynccnt(0);
  // Each lane reads only the LDS slot it wrote itself -> no cross-lane barrier needed.

  if (e >= E) return;

  const int uu = su[tid];
  const int vv = sv[tid];
  const int si = ss[tid];

  // Random gathers; the 800KB state row is L2-resident on MI455X (192MB L2).
  const float xu = Xt[uu];
  const float xv = Xt[vv];

  const float d  = xu - xv;
  const float kv = RHO_C + (1.0f - 2.0f * RHO_C) * fsig(SHARP_C * (eps - __builtin_fabsf(d)));
  kap[e] = kv;

  // Symmetric diffusion update; s in {0,1} -> skip ~half of all atomics.
  if (doUpd && si != 0) {
    const float dd = MU_C * (xv - xu);
    atomicAdd(Xn + uu,  dd);   // fire-and-forget global_atomic_add_f32
    atomicAdd(Xn + vv, -dd);
  }
}

extern "C" void kernel_launch(void* const* d_in, const int* in_sizes, int n_in,
                              void* d_out, int out_size, void* d_ws, size_t ws_size,
                              hipStream_t stream) {
  const float* logit = (const float*)d_in[0];
  const float* theta = (const float*)d_in[1];
  const int*   u     = (const int*)d_in[2];
  const int*   v     = (const int*)d_in[3];
  const int*   s     = (const int*)d_in[4];

  const int    N  = in_sizes[0];              // 200000
  const long   TE = (long)in_sizes[2];        // T*E
  int T = (int)(((long)out_size - TE) / (long)N);  // out = T*N + T*E
  if (T <= 0) T = 128;
  const int E = (int)(TE / T);                // 100000

  float* Xout = (float*)d_out;                         // [T, N]
  float* kap  = (float*)d_out + (size_t)T * (size_t)N; // [T*E]

  // Row 0: X0 = sigmoid(logit_X0)
  init_sigmoid_kernel<<<(N + 255) / 256, 256, 0, stream>>>(logit, Xout, N);

  const int egrid = (E + 255) / 256;
  for (int t = 0; t < T; ++t) {
    const int last = (t == T - 1);
    float* Xt = Xout + (size_t)t * N;
    float* Xn = last ? nullptr : (Xout + (size_t)(t + 1) * N);

    if (!last) {
      // Seed row t+1 with row t; stream order guarantees this completes
      // before the edge kernel's atomics touch row t+1.
      (void)hipMemcpyAsync(Xn, Xt, (size_t)N * sizeof(float),
                           hipMemcpyDeviceToDevice, stream);
    }

    edge_kernel<<<egrid, 256, 0, stream>>>(
        Xt, Xn,
        u + (size_t)t * E, v + (size_t)t * E, s + (size_t)t * E,
        theta, kap + (size_t)t * E, E, last ? 0 : 1);
  }
}